// CEMA_824633721546
// MI455X (gfx1250) — compile-verified
//
#include <hip/hip_runtime.h>
#include <stdint.h>

// EMA scan  ema_t = a*ema_{t-1} + (1-a)*x_t  over x[4, 4096, 2048] f32.
//
// Chunked-scan: per 16-step chunk,  Y = A_decay(16x16 lower-tri) * X(16x16 time x chan)
// via 4x V_WMMA_F32_16X16X4_F32 (K=4 each), then carry fixup
//   ema[t0+t] = alpha^(t+1)*C + Y[t],  carry broadcast across half-waves per chunk.
//
// Data movement: CDNA5 async copy.  Each wave owns a 16-chunk LDS ring (16 KB);
// GLOBAL_LOAD_ASYNC_TO_LDS_B128 (ASYNCcnt-tracked, no VGPR dest) keeps 32 async
// ops / 16 KB in flight per wave -> ~8 MB in flight device-wide (512 waves),
// enough MLP to approach the 23.3 TB/s HBM roofline.  Operands are then pulled
// from LDS with cheap ds_load_b32 in the WMMA B layout.

#define EMA_ALPHA 0.99f
#define EMA_OM    0.01f
#define D_MODEL   2048
#define SEQ_LEN   4096
#define BATCH     4
#define WAVES     4                // waves per block
#define NBUF      16               // LDS ring depth (16-step chunks) per wave
#define NCHUNK    (SEQ_LEN / 16)   // 256 chunks

typedef float v2f __attribute__((ext_vector_type(2)));
typedef float v8f __attribute__((ext_vector_type(8)));

__device__ __forceinline__ float apow(int e) {
    // alpha^e, e in [0, 16]
    float r = 1.0f, b = EMA_ALPHA;
    int n = e;
    while (n) { if (n & 1) r *= b; b *= b; n >>= 1; }
    return r;
}

// Enqueue one 16x16 f32 chunk (rows t..t+15, 16 channels) into the LDS ring.
// Lane l moves 16B of row (l>>2): 2 async B128 ops cover 8 rows x 64B each.
__device__ __forceinline__ void async_fill(const float* g0, uint32_t l0) {
    asm volatile("global_load_async_to_lds_b128 %0, %1, off"
                 :: "v"(l0), "v"(g0) : "memory");
    asm volatile("global_load_async_to_lds_b128 %0, %1, off"
                 :: "v"(l0 + 8 * 64), "v"(g0 + (size_t)8 * D_MODEL) : "memory");
}

__global__ __launch_bounds__(WAVES * 32) void ema_scan_wmma(
        const float* __restrict__ x, float* __restrict__ out) {
    __shared__ float ring[WAVES][NBUF][16][16];   // 64 KB per block

    const int lane = threadIdx.x & 31;
    const int wave = threadIdx.x >> 5;
    const int half = lane >> 4;    // 0: lanes 0-15, 1: lanes 16-31
    const int n    = lane & 15;    // channel within tile

    const int ch0 = (blockIdx.x * WAVES + wave) * 16;  // 16 channels per wave
    const int b   = blockIdx.y;

    const float* xb = x   + (size_t)b * SEQ_LEN * D_MODEL + ch0;
    float*       ob = out + (size_t)b * SEQ_LEN * D_MODEL + ch0 + n;

    // async-copy lane mapping: row r = lane>>2, 16B quarter cq = lane&3
    const int r  = lane >> 2;
    const int cq = lane & 3;
    const float*   gsrc     = xb + (size_t)r * D_MODEL + cq * 4;
    const uint32_t lds_lane = (uint32_t)(uintptr_t)&ring[wave][0][0][0]
                              + (uint32_t)(r * 64 + cq * 16);

    // A-matrix (16x4 f32 per WMMA): lane holds row m = n; component j holds
    // K = 4*kc + 2*half + j.  A[m][k] = (1-alpha)*alpha^(m-k) for k <= m else 0.
    v2f a[4];
#pragma unroll
    for (int kc = 0; kc < 4; ++kc) {
        const int k0 = 4 * kc + 2 * half;
        a[kc].x = (k0     <= n) ? EMA_OM * apow(n - k0)     : 0.0f;
        a[kc].y = (k0 + 1 <= n) ? EMA_OM * apow(n - k0 - 1) : 0.0f;
    }
    // carry scale per D register rr: alpha^(m+1), m = rr + 8*half
    float pscale[8];
#pragma unroll
    for (int rr = 0; rr < 8; ++rr) pscale[rr] = apow(rr + 8 * half + 1);

    // prologue: enqueue NBUF chunks (2 async B128 each -> 32 in flight)
    for (int i = 0; i < NBUF; ++i)
        async_fill(gsrc + (size_t)(i * 16) * D_MODEL, lds_lane + i * 1024);

    const float* bcol = &ring[wave][0][0][0] + n;   // column n of the ring
    float carry = 0.0f;
    int   slot  = 0;

    for (int i = 0; i < NCHUNK; ++i) {
        // retire the oldest chunk (async done returned in order)
        if (i < NCHUNK - NBUF) {
            asm volatile("s_wait_asynccnt 30" ::: "memory");
        } else if (i == NCHUNK - NBUF) {
            asm volatile("s_wait_asynccnt 0" ::: "memory");  // drain tail
        }

        // pull B operands from LDS in WMMA layout: lanes 0-15 rows {0,1} of
        // each K-group, lanes 16-31 rows {2,3}
        const float* sb = bcol + slot * 256;
        v2f tb[4];
#pragma unroll
        for (int kc = 0; kc < 4; ++kc) {
            const int k0 = 4 * kc + 2 * half;
            tb[kc].x = sb[(k0    ) * 16];
            tb[kc].y = sb[(k0 + 1) * 16];
        }

        // Y(16x16) = A * X, K accumulated over 4 WMMAs
        v8f c = {};
#pragma unroll
        for (int kc = 0; kc < 4; ++kc) {
            c = __builtin_amdgcn_wmma_f32_16x16x4_f32(
                /*neg_a=*/false, a[kc], /*neg_b=*/false, tb[kc],
                /*c_mod=*/(short)0, c, /*reuse_a=*/false, /*reuse_b=*/false);
        }

        // carry fixup + NT stores: D reg rr holds row m = rr + 8*half, col n
        const int t0 = i * 16;
        float last = 0.0f;
#pragma unroll
        for (int rr = 0; rr < 8; ++rr) {
            const float y = c[rr] + pscale[rr] * carry;
            __builtin_nontemporal_store(y, &ob[(size_t)(t0 + rr + 8 * half) * D_MODEL]);
            if (rr == 7) last = y;   // lanes 16-31 hold ema[t0+15]
        }
        carry = __shfl(last, 16 + n, 32);   // broadcast to both halves

        // refill this slot with chunk i+NBUF
        if (i + NBUF < NCHUNK)
            async_fill(gsrc + (size_t)((i + NBUF) * 16) * D_MODEL,
                       lds_lane + slot * 1024);

        slot = (slot + 1) & (NBUF - 1);
    }
}

extern "C" void kernel_launch(void* const* d_in, const int* in_sizes, int n_in,
                              void* d_out, int out_size, void* d_ws, size_t ws_size,
                              hipStream_t stream) {
    (void)in_sizes; (void)n_in; (void)out_size; (void)d_ws; (void)ws_size;
    const float* x = (const float*)d_in[0];
    float* out = (float*)d_out;

    dim3 block(WAVES * 32);                    // 4 waves (wave32)
    dim3 grid(D_MODEL / (16 * WAVES), BATCH);  // 16 ch/wave -> 512 waves total
    hipLaunchKernelGGL(ema_scan_wmma, grid, block, 0, stream, x, out);
}